// GatedGNN_15693810499780
// MI455X (gfx1250) — compile-verified
//
#include <hip/hip_runtime.h>
#include <hip/hip_bf16.h>

typedef float v2f __attribute__((ext_vector_type(2)));
typedef float v8f __attribute__((ext_vector_type(8)));

#define D 128
#define H3 384

__device__ __forceinline__ float sigmoidf_(float x) {
    return 1.0f / (1.0f + __expf(-x));
}

// ---------------------------------------------------------------------------
// flags: 3*N bytes in ws.  flags[t*N + v] = 1 iff node v has an incoming
// edge of type t.  segment_max of a dst-only function == masked copy.
// ---------------------------------------------------------------------------
__global__ void zero_words_kernel(int* __restrict__ p, int nwords) {
    int i = blockIdx.x * blockDim.x + threadIdx.x;
    if (i < nwords) p[i] = 0;
}

__global__ void mark_flags_kernel(const int* __restrict__ d0,
                                  const int* __restrict__ d1,
                                  const int* __restrict__ d2,
                                  unsigned char* __restrict__ flags,
                                  int N, int E) {
    int i = blockIdx.x * blockDim.x + threadIdx.x;
    if (i < E)            flags[d0[i]] = 1;
    else if (i < 2 * E)   flags[N + d1[i - E]] = 1;
    else if (i < 3 * E)   flags[2 * N + d2[i - 2 * E]] = 1;
}

// ---------------------------------------------------------------------------
// feats[v,:] = max_t( flag_t[v] ? node_state[v,:] @ W_t^T : 0 )
// One wave = one 16x16 tile.  8 waves/block cover all 8 column tiles of D=128.
// 3 accumulators share one A-fragment load per k-step.
// ---------------------------------------------------------------------------
__global__ __launch_bounds__(256) void feats_kernel(
    const float* __restrict__ ns,
    const float* __restrict__ Wa, const float* __restrict__ Wc,
    const float* __restrict__ Wd,
    const unsigned char* __restrict__ flags,
    float* __restrict__ feats, int N) {
    const int lane = threadIdx.x & 31;
    const int wave = threadIdx.x >> 5;
    const int row0 = blockIdx.x * 16;
    const int col0 = wave * 16;
    const int mn   = lane & 15;
    const int koff = (lane >> 4) * 2;

    const float* arow = ns + (size_t)(row0 + mn) * D + koff;
    const float* ba   = Wa + (size_t)(col0 + mn) * D + koff;
    const float* bc   = Wc + (size_t)(col0 + mn) * D + koff;
    const float* bd   = Wd + (size_t)(col0 + mn) * D + koff;

    v8f c0 = {}, c1 = {}, c2 = {};
#pragma unroll
    for (int kb = 0; kb < D; kb += 4) {
        v2f a  = *(const v2f*)(arow + kb);
        v2f b0 = *(const v2f*)(ba + kb);
        v2f b1 = *(const v2f*)(bc + kb);
        v2f b2 = *(const v2f*)(bd + kb);
        c0 = __builtin_amdgcn_wmma_f32_16x16x4_f32(false, a, false, b0, (short)0, c0, false, false);
        c1 = __builtin_amdgcn_wmma_f32_16x16x4_f32(false, a, false, b1, (short)0, c1, false, false);
        c2 = __builtin_amdgcn_wmma_f32_16x16x4_f32(false, a, false, b2, (short)0, c2, false, false);
    }
#pragma unroll
    for (int i = 0; i < 8; ++i) {
        int m   = i + ((lane >> 4) << 3);
        int row = row0 + m;
        float fa = flags[row]         ? c0[i] : 0.0f;
        float fc = flags[N + row]     ? c1[i] : 0.0f;
        float fd = flags[2 * N + row] ? c2[i] : 0.0f;
        feats[(size_t)row * D + col0 + mn] = fmaxf(fmaxf(fa, fc), fd);
    }
}

// ---------------------------------------------------------------------------
// gi = feats @ W_ih^T + b_ih   ([N,384]); grid.y covers 384 in chunks of 128.
// ---------------------------------------------------------------------------
__global__ __launch_bounds__(256) void gi_kernel(
    const float* __restrict__ feats, const float* __restrict__ Wih,
    const float* __restrict__ bih, float* __restrict__ gi) {
    const int lane = threadIdx.x & 31;
    const int wave = threadIdx.x >> 5;
    const int row0 = blockIdx.x * 16;
    const int col0 = blockIdx.y * 128 + wave * 16;
    const int mn   = lane & 15;
    const int koff = (lane >> 4) * 2;

    const float* arow = feats + (size_t)(row0 + mn) * D + koff;
    const float* brow = Wih   + (size_t)(col0 + mn) * D + koff;

    v8f c = {};
#pragma unroll
    for (int kb = 0; kb < D; kb += 4) {
        v2f a = *(const v2f*)(arow + kb);
        v2f b = *(const v2f*)(brow + kb);
        c = __builtin_amdgcn_wmma_f32_16x16x4_f32(false, a, false, b, (short)0, c, false, false);
    }
    float bias = bih[col0 + mn];
#pragma unroll
    for (int i = 0; i < 8; ++i) {
        int m = i + ((lane >> 4) << 3);
        gi[(size_t)(row0 + m) * H3 + col0 + mn] = c[i] + bias;
    }
}

// ---------------------------------------------------------------------------
// One GRU step.  Wave owns a 16x16 tile of h_out; needs the r/z/n column
// tiles of gh = h @ W_hh^T, i.e. W_hh rows col0, col0+128, col0+256.
// ---------------------------------------------------------------------------
__global__ __launch_bounds__(256) void gru_step_kernel(
    const float* __restrict__ hin, const float* __restrict__ gi,
    const float* __restrict__ Whh, const float* __restrict__ bhh,
    float* __restrict__ hout) {
    const int lane = threadIdx.x & 31;
    const int wave = threadIdx.x >> 5;
    const int row0 = blockIdx.x * 16;
    const int col0 = wave * 16;
    const int mn   = lane & 15;
    const int koff = (lane >> 4) * 2;

    const float* arow = hin + (size_t)(row0 + mn) * D + koff;
    const float* br   = Whh + (size_t)(col0 + mn) * D + koff;
    const float* bz   = Whh + (size_t)(128 + col0 + mn) * D + koff;
    const float* bn   = Whh + (size_t)(256 + col0 + mn) * D + koff;

    v8f cr = {}, cz = {}, cn = {};
#pragma unroll
    for (int kb = 0; kb < D; kb += 4) {
        v2f a  = *(const v2f*)(arow + kb);
        v2f b0 = *(const v2f*)(br + kb);
        v2f b1 = *(const v2f*)(bz + kb);
        v2f b2 = *(const v2f*)(bn + kb);
        cr = __builtin_amdgcn_wmma_f32_16x16x4_f32(false, a, false, b0, (short)0, cr, false, false);
        cz = __builtin_amdgcn_wmma_f32_16x16x4_f32(false, a, false, b1, (short)0, cz, false, false);
        cn = __builtin_amdgcn_wmma_f32_16x16x4_f32(false, a, false, b2, (short)0, cn, false, false);
    }
    const float bhr = bhh[col0 + mn];
    const float bhz = bhh[128 + col0 + mn];
    const float bhn = bhh[256 + col0 + mn];
#pragma unroll
    for (int i = 0; i < 8; ++i) {
        int m   = i + ((lane >> 4) << 3);
        int row = row0 + m;
        size_t gb = (size_t)row * H3 + col0 + mn;
        float r  = sigmoidf_(gi[gb]       + cr[i] + bhr);
        float z  = sigmoidf_(gi[gb + 128] + cz[i] + bhz);
        float ng = tanhf(gi[gb + 256] + r * (cn[i] + bhn));
        float h  = hin[(size_t)row * D + col0 + mn];
        hout[(size_t)row * D + col0 + mn] = (1.0f - z) * ng + z * h;
    }
}

// ---------------------------------------------------------------------------
extern "C" void kernel_launch(void* const* d_in, const int* in_sizes, int n_in,
                              void* d_out, int out_size, void* d_ws, size_t ws_size,
                              hipStream_t stream) {
    const float* ns   = (const float*)d_in[0];
    const int*   da   = (const int*)d_in[1];
    const int*   dc   = (const int*)d_in[2];
    const int*   dd   = (const int*)d_in[3];
    const float* Wa   = (const float*)d_in[4];
    const float* Wc   = (const float*)d_in[5];
    const float* Wd   = (const float*)d_in[6];
    const float* Wih  = (const float*)d_in[7];
    const float* Whh  = (const float*)d_in[8];
    const float* bih  = (const float*)d_in[9];
    const float* bhh  = (const float*)d_in[10];
    float* out = (float*)d_out;

    const int N = in_sizes[0] / D;   // 50000 (== 3125 * 16, no remainder)
    const int E = in_sizes[1];       // 800000

    char* ws = (char*)d_ws;
    float* gi    = (float*)ws;                                   // N*384 f32
    float* feats = (float*)(ws + (size_t)N * H3 * sizeof(float)); // N*128 f32 (reused as h buffer)
    unsigned char* flags = (unsigned char*)(ws + (size_t)N * (H3 + D) * sizeof(float)); // 3*N bytes

    // 1) build per-node incidence flags
    int nwords = (3 * N + 3) / 4;
    zero_words_kernel<<<(nwords + 255) / 256, 256, 0, stream>>>((int*)flags, nwords);
    mark_flags_kernel<<<(3 * E + 255) / 256, 256, 0, stream>>>(da, dc, dd, flags, N, E);

    const int rowBlocks = N / 16;  // 3125

    // 2) feats = max_t( flag_t ? ns @ W_t^T : 0 )
    feats_kernel<<<rowBlocks, 256, 0, stream>>>(ns, Wa, Wc, Wd, flags, feats, N);

    // 3) gi = feats @ W_ih^T + b_ih
    {
        dim3 g(rowBlocks, H3 / 128);
        gi_kernel<<<g, 256, 0, stream>>>(feats, Wih, bih, gi);
    }

    // 4) four GRU steps; feats buffer is dead now, reuse as h ping-pong
    gru_step_kernel<<<rowBlocks, 256, 0, stream>>>(ns,    gi, Whh, bhh, feats); // h1
    gru_step_kernel<<<rowBlocks, 256, 0, stream>>>(feats, gi, Whh, bhh, out);   // h2
    gru_step_kernel<<<rowBlocks, 256, 0, stream>>>(out,   gi, Whh, bhh, feats); // h3
    gru_step_kernel<<<rowBlocks, 256, 0, stream>>>(feats, gi, Whh, bhh, out);   // h4
}